// StationarityRouter_71073118814442
// MI455X (gfx1250) — compile-verified
//
#include <hip/hip_runtime.h>
#include <hip/hip_bf16.h>
#include <math.h>

// ---------------------------------------------------------------------------
// StationarityRouter fused kernels for gfx1250 (MI455X, wave32)
//   Kernel 1: one workgroup per row (L=1024). Single streaming read of x
//             (nontemporal b128 loads), all features computed from LDS,
//             radix-2 FFT in LDS, writes feats_n[B,9] (layernormed).
//   Kernel 2: MLP 9->64->3 via V_WMMA_F32_16X16X4_F32 (f32 WMMA), softmax,
//             writes weights[B,3].
// d_out layout: [weights B*3][feats_n B*9]
// ---------------------------------------------------------------------------

typedef __attribute__((ext_vector_type(2))) float v2f;
typedef __attribute__((ext_vector_type(4))) float v4f;
typedef __attribute__((ext_vector_type(8))) float v8f;

#define ROW_L 1024
#define NT 256

__device__ __forceinline__ float block_sum(float v, float* red, int tid) {
  __syncthreads();            // protect previous use of red[]
  red[tid] = v;
  __syncthreads();
#pragma unroll
  for (int s = NT / 2; s > 0; s >>= 1) {
    if (tid < s) red[tid] += red[tid + s];
    __syncthreads();
  }
  return red[0];
}

__device__ __forceinline__ float block_max(float v, float* red, int tid) {
  __syncthreads();
  red[tid] = v;
  __syncthreads();
#pragma unroll
  for (int s = NT / 2; s > 0; s >>= 1) {
    if (tid < s) red[tid] = fmaxf(red[tid], red[tid + s]);
    __syncthreads();
  }
  return red[0];
}

__device__ __forceinline__ float clamp11(float v) { return fminf(fmaxf(v, -1.0f), 1.0f); }
__device__ __forceinline__ float clamp01(float v) { return fminf(fmaxf(v, 0.0f), 1.0f); }

__global__ void __launch_bounds__(NT)
feats_kernel(const float* __restrict__ x,
             const float* __restrict__ ln_w,
             const float* __restrict__ ln_b,
             float* __restrict__ feats_out /* [B,9] layernormed */) {
  __shared__ float xs[ROW_L];     // row copy, later reused for PSD bins 0..512
  __shared__ float wa[ROW_L];     // FFT real
  __shared__ float wb[ROW_L];     // FFT imag
  __shared__ float twr[ROW_L / 2];
  __shared__ float twi[ROW_L / 2];
  __shared__ float red[NT];
  __shared__ float sscan[NT];

  const int tid = threadIdx.x;
  const int row = blockIdx.x;
  const float* xr = x + (size_t)row * ROW_L;

  // --- load row (nontemporal streaming, 16B per lane, coalesced) ---
  {
    v4f v = __builtin_nontemporal_load((const v4f*)xr + tid);
    ((v4f*)xs)[tid] = v;
  }
  // --- precompute FFT twiddles w_j = exp(-2*pi*i*j/1024), j=0..511 ---
  for (int j = tid; j < ROW_L / 2; j += NT) {
    float s_, c_;
    __sincosf(-6.283185307179586f * (float)j * (1.0f / (float)ROW_L), &s_, &c_);
    twr[j] = c_;
    twi[j] = s_;
  }
  __syncthreads();

  const int i0 = tid * 4;
  const float xlast = xs[ROW_L - 1];

  // ================= Phase A: raw moments, lag1(raw), diff, trend =========
  float s1 = 0.f, s2 = 0.f, p1 = 0.f, sd = 0.f, sd2 = 0.f, st = 0.f, st2 = 0.f;
  float lres[4];
#pragma unroll
  for (int j = 0; j < 4; ++j) {
    int i = i0 + j;
    float xi = xs[i];
    s1 += xi;
    s2 += xi * xi;
    if (i >= 1) p1 += xi * xs[i - 1];
    if (i < ROW_L - 1) {
      float d = xs[i + 1] - xi;
      sd += d;
      sd2 += d * d;
    }
    // moving average k=25 with edge replication
    float acc = 0.f;
#pragma unroll
    for (int dd = -12; dd <= 12; ++dd) {
      int idx = i + dd;
      idx = idx < 0 ? 0 : (idx > ROW_L - 1 ? ROW_L - 1 : idx);
      acc += xs[idx];
    }
    float tr = acc * (1.0f / 25.0f);
    st += tr;
    st2 += tr * tr;
    lres[j] = xi - tr;  // residual
  }

  // --- block scan over residuals for cumsum variance (KPSS proxy) ---
  float l0 = lres[0];
  float l1 = l0 + lres[1];
  float l2 = l1 + lres[2];
  float l3 = l2 + lres[3];
  __syncthreads();
  sscan[tid] = l3;
  __syncthreads();
  for (int off = 1; off < NT; off <<= 1) {
    float v = sscan[tid];
    float add = (tid >= off) ? sscan[tid - off] : 0.0f;
    __syncthreads();
    sscan[tid] = v + add;
    __syncthreads();
  }
  float excl = (tid == 0) ? 0.0f : sscan[tid - 1];
  float c0 = excl + l0, c1 = excl + l1, c2 = excl + l2, c3 = excl + l3;
  float sc = c0 + c1 + c2 + c3;
  float sc2 = c0 * c0 + c1 * c1 + c2 * c2 + c3 * c3;

  float S1 = block_sum(s1, red, tid);
  float S2 = block_sum(s2, red, tid);
  float P1 = block_sum(p1, red, tid);
  float Sd = block_sum(sd, red, tid);
  float Sd2 = block_sum(sd2, red, tid);
  float St = block_sum(st, red, tid);
  float St2 = block_sum(st2, red, tid);
  float Sc = block_sum(sc, red, tid);
  float Sc2 = block_sum(sc2, red, tid);

  // ================= Phase B: centered lag products (ACF 1..10) ===========
  const float m = S1 * (1.0f / (float)ROW_L);
  float q = 0.f;
  float ck[10];
#pragma unroll
  for (int k = 0; k < 10; ++k) ck[k] = 0.f;
#pragma unroll
  for (int j = 0; j < 4; ++j) {
    int i = i0 + j;
    float xc = xs[i] - m;
    q += xc * xc;
#pragma unroll
    for (int k = 1; k <= 10; ++k) {
      if (i + k < ROW_L) ck[k - 1] += xc * (xs[i + k] - m);
    }
  }
  float Q = block_sum(q, red, tid);
  float Cr[10];
  for (int k = 0; k < 10; ++k) Cr[k] = block_sum(ck[k], red, tid);

  // ================= Phase C: 1024-pt radix-2 FFT in LDS ==================
  // bit-reversed load: wa/wb <- xs
#pragma unroll
  for (int j = 0; j < 4; ++j) {
    int i = i0 + j;
    int r = (int)(__brev((unsigned)i) >> 22);
    wa[i] = xs[r];
    wb[i] = 0.0f;
  }
  __syncthreads();
#pragma unroll
  for (int s = 1; s <= 10; ++s) {
    int mlen = 1 << s, half = mlen >> 1;
#pragma unroll
    for (int t = 0; t < 2; ++t) {
      int b = tid + t * NT;  // butterfly 0..511
      int pos = b & (half - 1);
      int blk = b >> (s - 1);
      int k = blk * mlen + pos;
      int tw = pos << (10 - s);
      float wr = twr[tw], wi = twi[tw];
      float xr2 = wa[k + half], xi2 = wb[k + half];
      float trr = wr * xr2 - wi * xi2;
      float tri = wr * xi2 + wi * xr2;
      float ur = wa[k], ui = wb[k];
      wa[k] = ur + trr;
      wb[k] = ui + tri;
      wa[k + half] = ur - trr;
      wb[k + half] = ui - tri;
    }
    __syncthreads();
  }

  // --- PSD bins 0..512 (xs reused as psd storage) ---
  float psum = 0.f, pmax = 0.f;
#pragma unroll
  for (int t = 0; t < 2; ++t) {
    int k = tid + t * NT;  // 0..511
    float pr = wa[k], pi = wb[k];
    float ps = fmaxf(pr * pr + pi * pi, 1e-8f);
    xs[k] = ps;
    psum += ps;
    pmax = fmaxf(pmax, ps);
  }
  if (tid == 0) {
    float pr = wa[512], pi = wb[512];
    float ps = fmaxf(pr * pr + pi * pi, 1e-8f);
    xs[512] = ps;
    psum += ps;
    pmax = fmaxf(pmax, ps);
  }
  float PSUM = block_sum(psum, red, tid);
  float PMAX = block_max(pmax, red, tid);

  float entp = 0.f;
  float invps = 1.0f / PSUM;
#pragma unroll
  for (int t = 0; t < 2; ++t) {
    int k = tid + t * NT;
    float pn = xs[k] * invps;
    entp -= pn * __logf(pn + 1e-8f);
  }
  if (tid == 0) {
    float pn = xs[512] * invps;
    entp -= pn * __logf(pn + 1e-8f);
  }
  float ENT = block_sum(entp, red, tid);

  // ================= Phase D: features + layernorm ========================
  if (tid == 0) {
    float f[9];
    float den = Q + 1e-8f;
    float invden = 1.0f / den;
    float lbq = 0.f;
    float acfv[10];
#pragma unroll
    for (int k = 0; k < 10; ++k) {
      acfv[k] = clamp11(Cr[k] * invden);
      lbq += acfv[k] * acfv[k];
    }
    f[0] = acfv[0];
    f[1] = acfv[1];
    float max_ent = __logf(513.0f + 1e-8f);
    f[2] = clamp01(ENT / (max_ent + 1e-8f));
    f[3] = clamp11(P1 / (S2 - xlast * xlast + 1e-8f));
    {
      float nInv = 1.0f / 1023.0f;
      float mu = Sd * nInv;
      f[4] = Sd2 * nInv - mu * mu;  // diff_var
    }
    {
      float nInv = 1.0f / (float)ROW_L;
      float mu = St * nInv;
      f[5] = St2 * nInv - mu * mu;  // trend_strength
      float muc = Sc * nInv;
      float cvar = Sc2 * nInv - muc * muc;
      float xvar = Q * nInv;
      f[6] = cvar / (xvar + 1e-8f);  // kpss_proxy
    }
    f[7] = lbq;
    f[8] = clamp01(PMAX / (PSUM + 1e-8f));

    float fm = 0.f;
#pragma unroll
    for (int d = 0; d < 9; ++d) fm += f[d];
    fm *= (1.0f / 9.0f);
    float fv = 0.f;
#pragma unroll
    for (int d = 0; d < 9; ++d) {
      float dd = f[d] - fm;
      fv += dd * dd;
    }
    fv *= (1.0f / 9.0f);
    float inv = rsqrtf(fv + 1e-5f);
    float* o = feats_out + (size_t)row * 9;
#pragma unroll
    for (int d = 0; d < 9; ++d) o[d] = (f[d] - fm) * inv * ln_w[d] + ln_b[d];
  }
}

// ---------------------------------------------------------------------------
// Kernel 2: MLP via V_WMMA_F32_16X16X4_F32.
// Block = 128 threads (4 waves), 16 rows per block.
// GEMM1: per-wave 16x16 tile of hidden (N=64), K=9 padded to 12 -> 3 WMMAs.
// GEMM2: wave 0, K=64 -> 16 WMMAs, N=3 padded to 16. Softmax on 16 threads.
// ---------------------------------------------------------------------------
__global__ void __launch_bounds__(128)
mlp_kernel(const float* __restrict__ feats, const float* __restrict__ w1,
           const float* __restrict__ b1, const float* __restrict__ w2,
           const float* __restrict__ b2, float* __restrict__ weights) {
  __shared__ float sf[16 * 12];   // feats tile, K padded 9->12
  __shared__ float sw1[12 * 64];  // w1 padded
  __shared__ float sh[16 * 64];   // hidden (post bias+relu)
  __shared__ float sw2[64 * 16];  // w2 padded N 3->16
  __shared__ float slog[16 * 16]; // logits (n<3 valid)

  const int tid = threadIdx.x;
  const int lane = tid & 31;
  const int wave = tid >> 5;
  const int rowbase = blockIdx.x * 16;

  for (int idx = tid; idx < 16 * 12; idx += 128) {
    int r = idx / 12, c = idx - r * 12;
    sf[idx] = (c < 9) ? feats[(size_t)(rowbase + r) * 9 + c] : 0.0f;
  }
  for (int idx = tid; idx < 12 * 64; idx += 128) {
    int k = idx >> 6, n = idx & 63;
    sw1[idx] = (k < 9) ? w1[k * 64 + n] : 0.0f;
  }
  for (int idx = tid; idx < 64 * 16; idx += 128) {
    int k = idx >> 4, n = idx & 15;
    sw2[idx] = (n < 3) ? w2[k * 3 + n] : 0.0f;
  }
  __syncthreads();

  const int mrow = lane & 15;          // A: M index
  const int kb = (lane >> 4) * 2;      // A/B: K sub-offset per half-wave
  const int ncol = (lane & 15);        // B/C/D: N index within tile

  // ---- GEMM1: hidden tile for this wave ----
  v8f acc = {};
#pragma unroll
  for (int step = 0; step < 3; ++step) {
    int k0 = step * 4 + kb;
    v2f a, b;
    a.x = sf[mrow * 12 + k0];
    a.y = sf[mrow * 12 + k0 + 1];
    b.x = sw1[k0 * 64 + 16 * wave + ncol];
    b.y = sw1[(k0 + 1) * 64 + 16 * wave + ncol];
    acc = __builtin_amdgcn_wmma_f32_16x16x4_f32(false, a, false, b, (short)0,
                                                acc, false, false);
  }
  {
    int mofs = (lane >> 4) * 8;
    float bias = b1[16 * wave + ncol];
#pragma unroll
    for (int r = 0; r < 8; ++r) {
      float h = acc[r] + bias;
      sh[(mofs + r) * 64 + 16 * wave + ncol] = fmaxf(h, 0.0f);
    }
  }
  __syncthreads();

  // ---- GEMM2: logits (wave 0 only; EXEC all-1s within the wave) ----
  if (wave == 0) {
    v8f acc2 = {};
#pragma unroll
    for (int step = 0; step < 16; ++step) {
      int k0 = step * 4 + kb;
      v2f a, b;
      a.x = sh[mrow * 64 + k0];
      a.y = sh[mrow * 64 + k0 + 1];
      b.x = sw2[k0 * 16 + ncol];
      b.y = sw2[(k0 + 1) * 16 + ncol];
      acc2 = __builtin_amdgcn_wmma_f32_16x16x4_f32(false, a, false, b, (short)0,
                                                   acc2, false, false);
    }
    int mofs = (lane >> 4) * 8;
#pragma unroll
    for (int r = 0; r < 8; ++r) slog[(mofs + r) * 16 + ncol] = acc2[r];
  }
  __syncthreads();

  // ---- bias + softmax over 3 classes, one thread per row ----
  if (tid < 16) {
    float z0 = slog[tid * 16 + 0] + b2[0];
    float z1 = slog[tid * 16 + 1] + b2[1];
    float z2 = slog[tid * 16 + 2] + b2[2];
    float mx = fmaxf(z0, fmaxf(z1, z2));
    float e0 = __expf(z0 - mx), e1 = __expf(z1 - mx), e2 = __expf(z2 - mx);
    float inv = 1.0f / (e0 + e1 + e2);
    float* o = weights + (size_t)(rowbase + tid) * 3;
    o[0] = e0 * inv;
    o[1] = e1 * inv;
    o[2] = e2 * inv;
  }
}

extern "C" void kernel_launch(void* const* d_in, const int* in_sizes, int n_in,
                              void* d_out, int out_size, void* d_ws, size_t ws_size,
                              hipStream_t stream) {
  const float* x    = (const float*)d_in[0];
  const float* ln_w = (const float*)d_in[1];
  const float* ln_b = (const float*)d_in[2];
  const float* w1   = (const float*)d_in[3];
  const float* b1   = (const float*)d_in[4];
  const float* w2   = (const float*)d_in[5];
  const float* b2   = (const float*)d_in[6];

  const int B = in_sizes[0] / ROW_L;  // 32768
  float* out = (float*)d_out;
  float* weights = out;                     // [B,3]
  float* feats_n = out + (size_t)B * 3;     // [B,9]

  feats_kernel<<<B, NT, 0, stream>>>(x, ln_w, ln_b, feats_n);
  mlp_kernel<<<B / 16, 128, 0, stream>>>(feats_n, w1, b1, w2, b2, weights);
}